// GatedGCNNet_25632364822814
// MI455X (gfx1250) — compile-verified
//
#include <hip/hip_runtime.h>

typedef __attribute__((ext_vector_type(2))) float v2f;
typedef __attribute__((ext_vector_type(8))) float v8f;

constexpr int  BB = 64, NN = 64, DD = 64;
constexpr int  BN = BB * NN;                    // 4096 node rows
constexpr long ND = (long)BN * DD;              // 262144
constexpr long NE = (long)BB * NN * NN * DD;    // 16777216 (one [B,N,N,D] tensor)

// ---------------------------------------------------------------------------
// Generic batched WMMA GEMM:  C = act(alpha * op(A)·op(B) + bias)
// One wave per 16x16 tile; K = 4*KS fully unrolled: burst-load all fragments,
// then KS back-to-back V_WMMA_F32_16X16X4_F32 on a chained accumulator.
// TA/TB: transpose flags (compile time).  ACT: 0 none, 1 relu, 2 elu+1.
// NT: non-temporal C store (single-use streams; keep reused tensors in L2).
// ---------------------------------------------------------------------------
template <int TA, int TB, int ACT, int KS, int NT>
__global__ void wmma_gemm_kernel(const float* __restrict__ A, const float* __restrict__ B,
                                 const float* __restrict__ bias, float* __restrict__ C,
                                 int lda, int ldb, int ldc,
                                 long sA, long sB, long sC, float alpha)
{
  int lane = threadIdx.x;          // 0..31 (wave32)
  int half = lane >> 4, l16 = lane & 15;
  int  n0 = blockIdx.x * 16;
  long m0 = (long)blockIdx.y * 16;
  const float* Ab = A + (long)blockIdx.z * sA;
  const float* Bb = B + (long)blockIdx.z * sB;
  float*       Cb = C + (long)blockIdx.z * sC;
  long ar = m0 + l16;              // A row (M)
  int  bc = n0 + l16;              // B col (N)

  v2f af[KS], bf[KS];
#pragma unroll
  for (int s = 0; s < KS; ++s) {   // lanes 0-15: K=4s,4s+1 ; lanes 16-31: K=4s+2,4s+3
    int ka = 4 * s + 2 * half;
    if (TA) { af[s].x = Ab[(long)ka * lda + ar]; af[s].y = Ab[(long)(ka + 1) * lda + ar]; }
    else    { af[s] = *(const v2f*)(Ab + ar * lda + ka); }          // contiguous pair
    if (TB) { bf[s] = *(const v2f*)(Bb + (long)bc * ldb + ka); }    // contiguous pair
    else    { bf[s].x = Bb[(long)ka * ldb + bc]; bf[s].y = Bb[(long)(ka + 1) * ldb + bc]; }
  }
  v8f acc = {};
#pragma unroll
  for (int s = 0; s < KS; ++s)
    acc = __builtin_amdgcn_wmma_f32_16x16x4_f32(false, af[s], false, bf[s],
                                                (short)0, acc, false, false);

  float bv = bias ? bias[bc] : 0.f;
#pragma unroll
  for (int r = 0; r < 8; ++r) {    // C/D: VGPR r -> rows r (lanes 0-15) and r+8 (lanes 16-31)
    long row = m0 + r + 8 * half;
    float v = acc[r] * alpha + bv;
    if (ACT == 1)      v = fmaxf(v, 0.f);
    else if (ACT == 2) v = (v > 0.f) ? (v + 1.f) : __expf(v);       // elu(x)+1
    if (NT) __builtin_nontemporal_store(v, &Cb[row * ldc + bc]);
    else    Cb[row * ldc + bc] = v;
  }
}

template <int TA, int TB, int ACT, int KS, int NT = 0>
static void gemm(hipStream_t st, const float* A, const float* B, const float* bias, float* C,
                 long M, int N, int batches = 1, long sA = 0, long sB = 0, long sC = 0,
                 float alpha = 1.f)
{
  constexpr int K = 4 * KS;
  int lda = TA ? (int)M : K;       // TA: A stored [K x M]
  int ldb = TB ? K : N;            // TB: B stored [N x K]
  dim3 g((unsigned)(N / 16), (unsigned)(M / 16), (unsigned)batches);
  wmma_gemm_kernel<TA, TB, ACT, KS, NT><<<g, dim3(32), 0, st>>>(
      A, B, bias, C, lda, ldb, N, sA, sB, sC, alpha);
}

// ---------------------------------------------------------------------------
// Elementwise / reduction kernels
// ---------------------------------------------------------------------------
__global__ void kzero(float* p, int n) {
  int i = blockIdx.x * blockDim.x + threadIdx.x;
  if (i < n) p[i] = 0.f;
}
__global__ void kcopy(const float* a, float* b, long n) {
  long i = (long)blockIdx.x * blockDim.x + threadIdx.x;
  if (i < n) b[i] = a[i];
}

// BN over nodes per (b, channel), then ReLU, in place.  x: [B, NN, Cn]
__global__ void kbn_nodes(float* x, int Cn, int total) {
  int idx = blockIdx.x * blockDim.x + threadIdx.x;
  if (idx >= total) return;
  int b = idx / Cn, c = idx % Cn;
  long base = (long)b * NN * Cn + c;
  float s = 0.f, q = 0.f;
  for (int n = 0; n < NN; ++n) { float v = x[base + (long)n * Cn]; s += v; q += v * v; }
  float mu = s / NN, var = q / NN - mu * mu;
  float rs = rsqrtf(var + 1e-5f);
  for (int n = 0; n < NN; ++n) {
    long ix = base + (long)n * Cn;
    x[ix] = fmaxf((x[ix] - mu) * rs, 0.f);
  }
}

// y1[b,n,h'] = sum_h Wpg[h] * x1[b,n,h*64+h']
__global__ void ky1(const float* __restrict__ x1, const float* __restrict__ Wpg, float* y1) {
  int idx = blockIdx.x * blockDim.x + threadIdx.x;
  if (idx >= (int)ND) return;
  int hp = idx & 63, nd = idx >> 6;
  float s = 0.f;
  for (int h = 0; h < 64; ++h) s += Wpg[h] * x1[(long)nd * 4096 + h * 64 + hp];
  y1[idx] = s;
}

// zero diag, dual softmax product, edge threshold OR adjacency  (one block per graph)
__global__ void kgtp_mask(const float* __restrict__ mmv, const float* bpg,
                          const int* __restrict__ adj, float* maskf) {
  __shared__ float m[64][64];
  __shared__ float cmax[64], csum[64], rmax[64], rsum[64];
  int b = blockIdx.x, t = threadIdx.x;
  float bp = bpg[0];
  for (int j = 0; j < 64; ++j)
    m[t][j] = (t == j) ? 0.f : (mmv[((long)b * 64 + t) * 64 + j] + bp);
  __syncthreads();
  float mx = -1e30f;                               // column t: softmax over i (axis=1)
  for (int i = 0; i < 64; ++i) mx = fmaxf(mx, m[i][t]);
  float s = 0.f;
  for (int i = 0; i < 64; ++i) s += __expf(m[i][t] - mx);
  cmax[t] = mx; csum[t] = s;
  mx = -1e30f;                                     // row t: softmax over j (axis=2)
  for (int j = 0; j < 64; ++j) mx = fmaxf(mx, m[t][j]);
  s = 0.f;
  for (int j = 0; j < 64; ++j) s += __expf(m[t][j] - mx);
  rmax[t] = mx; rsum[t] = s;
  __syncthreads();
  for (int j = 0; j < 64; ++j) {
    float p = (__expf(m[t][j] - cmax[j]) / csum[j]) * (__expf(m[t][j] - rmax[t]) / rsum[t]);
    int a = adj[((long)b * 64 + t) * 64 + j];
    maskf[((long)b * 64 + t) * 64 + j] = (a > 0 || p > 0.3f) ? 1.f : 0.f;
  }
}

__global__ void kcount(const float* maskf, float* cnt, int n) {
  __shared__ float sm[256];
  int idx = blockIdx.x * 256 + threadIdx.x;
  sm[threadIdx.x] = (idx < n) ? maskf[idx] : 0.f;
  __syncthreads();
  for (int s = 128; s > 0; s >>= 1) {
    if (threadIdx.x < s) sm[threadIdx.x] += sm[threadIdx.x + s];
    __syncthreads();
  }
  if (threadIdx.x == 0) atomicAdd(cnt, sm[0]);
}

// per-node Conv1d halves (src weights cw[0..2], dst weights cw[3..5])
__global__ void kconv(const float* __restrict__ feat, const float* __restrict__ cw,
                      float* convS, float* convD) {
  int idx = blockIdx.x * blockDim.x + threadIdx.x;
  if (idx >= (int)ND) return;
  int d = idx & 63, nd = idx >> 6;
  const float* fr = feat + (long)nd * 64;
  float fm = d > 0 ? fr[d - 1] : 0.f;
  float f0 = fr[d];
  float fp = d < 63 ? fr[d + 1] : 0.f;
  convS[idx] = cw[0] * fm + cw[1] * f0 + cw[2] * fp;
  convD[idx] = cw[3] * fm + cw[4] * f0 + cw[5] * fp;
}

// econ[d] = 1 (e_ones) + conv_b * colsum(Wp2)[d] + bp2[d]
__global__ void keconst(const float* Wp2, const float* bp2, const float* cb, float* ec) {
  int d = threadIdx.x;
  float s = 0.f;
  for (int k = 0; k < 64; ++k) s += Wp2[k * 64 + d];
  ec[d] = 1.f + cb[0] * s + bp2[d];
}
// cC[d] = colsum(CW)[d] + Cb[d]   (Ce for e == ones)
__global__ void kcc(const float* CW, const float* Cb, float* cC) {
  int d = threadIdx.x;
  float s = 0.f;
  for (int k = 0; k < 64; ++k) s += CW[k * 64 + d];
  cC[d] = s + Cb[d];
}

// GatedGCN layer with e == ones (no edge output).  One block per (b,j), thread = d.
__global__ void kgcn_light(const float* __restrict__ hA, const float* __restrict__ hB,
                           const float* __restrict__ hD, const float* __restrict__ hE,
                           const float* __restrict__ cC, const float* __restrict__ maskf,
                           float* tmp, float* ssum, float* ssq) {
  int bj = blockIdx.x, b = bj >> 6, j = bj & 63, d = threadIdx.x;
  float Ej = hE[(long)bj * 64 + d] + cC[d];
  float num = 0.f, den = 1e-6f;
  for (int i = 0; i < 64; ++i) {
    float en = hD[((long)b * 64 + i) * 64 + d] + Ej;
    float s = (1.f / (1.f + __expf(-en))) * maskf[((long)b * 64 + i) * 64 + j];
    num += s * hB[((long)b * 64 + i) * 64 + d];
    den += s;
  }
  float t = hA[(long)bj * 64 + d] + num / den;
  tmp[(long)bj * 64 + d] = t;
  atomicAdd(&ssum[d], t);
  atomicAdd(&ssq[d], t * t);
}

// h += relu(BN_{(0,1)}(tmp))
__global__ void kbn_node_fin(float* h, const float* tmp, const float* ssum, const float* ssq,
                             float inv_n) {
  int idx = blockIdx.x * blockDim.x + threadIdx.x;
  if (idx >= (int)ND) return;
  int d = idx & 63;
  float mu  = ssum[d] * inv_n;
  float var = ssq[d] * inv_n - mu * mu;
  float v = (tmp[idx] - mu) * rsqrtf(var + 1e-5f);
  h[idx] += fmaxf(v, 0.f);
}

// out[b,d] = scale * sum_n x[b,n,d]
__global__ void ksum_nodes(const float* x, float* out, float scale) {
  int b = blockIdx.x, d = threadIdx.x;
  float s = 0.f;
  for (int n = 0; n < 64; ++n) s += x[((long)b * 64 + n) * 64 + d];
  out[b * 64 + d] = s * scale;
}

// FAM linear attention (src rows all equal): msg = coef * v,  coef = N*s/(N*s+eps)
__global__ void kfam_msg(const float* __restrict__ qq, const float* __restrict__ Kb,
                         const float* __restrict__ vb, float* msg) {
  int idx = blockIdx.x * blockDim.x + threadIdx.x;
  if (idx >= BN) return;
  int b = idx >> 6;
  float s = 0.f;
  for (int d = 0; d < 64; ++d) {
    float q = qq[(long)idx * 64 + d];
    q = (q > 0.f) ? (q + 1.f) : __expf(q);
    s += q * Kb[b * 64 + d];
  }
  float coef = 64.f * s / (64.f * s + 1e-6f);
  for (int d = 0; d < 64; ++d) msg[(long)idx * 64 + d] = coef * vb[b * 64 + d];
}

// ARM linear attention (pooled over i): msg_j = (Q_j @ KV) * 4096 / (Q_j.Ksum + eps)
__global__ void karm_msg(const float* __restrict__ q2, const float* __restrict__ Ksum,
                         const float* __restrict__ KV, float* msg) {
  __shared__ float Q[64];
  int bj = blockIdx.x, b = bj >> 6, t = threadIdx.x;
  float q = q2[(long)bj * 64 + t];
  Q[t] = (q > 0.f) ? (q + 1.f) : __expf(q);
  __syncthreads();
  float dot = 0.f;
  for (int d = 0; d < 64; ++d) dot += Q[d] * Ksum[b * 64 + d];
  float z = 4096.f / (dot + 1e-6f);
  float acc = 0.f;
  for (int d = 0; d < 64; ++d) acc += Q[d] * KV[((long)b * 64 + d) * 64 + t];
  msg[(long)bj * 64 + t] = acc * z;
}

// LayerNorm rows of 64 (+ optional residual)
__global__ void kln(const float* __restrict__ x, const float* g, const float* bt,
                    const float* res, float* out, int rows) {
  int r = blockIdx.x * blockDim.x + threadIdx.x;
  if (r >= rows) return;
  const float* xr = x + (long)r * 64;
  float s = 0.f;
  for (int d = 0; d < 64; ++d) s += xr[d];
  float mu = s / 64.f, q = 0.f;
  for (int d = 0; d < 64; ++d) { float t = xr[d] - mu; q += t * t; }
  float rs = rsqrtf(q / 64.f + 1e-5f);
  float* orow = out + (long)r * 64;
  const float* rr = res ? res + (long)r * 64 : nullptr;
  for (int d = 0; d < 64; ++d) {
    float v = (xr[d] - mu) * rs * g[d] + bt[d];
    orow[d] = rr ? rr[d] + v : v;
  }
}

__global__ void kconcat(const float* a, const float* b, float* cat, int rows) {
  int idx = blockIdx.x * blockDim.x + threadIdx.x;
  if (idx >= rows * 128) return;
  int r = idx >> 7, c = idx & 127;
  cat[idx] = (c < 64) ? a[(long)r * 64 + c] : b[(long)r * 64 + c - 64];
}

__global__ void kaddvec3(const float* Qn, const float* Rn, const float* ec, float* QR) {
  int idx = blockIdx.x * blockDim.x + threadIdx.x;
  if (idx >= (int)ND) return;
  QR[idx] = Qn[idx] + Rn[idx] + ec[idx & 63];
}

// e_init = P_i + QR_j, accumulate masked sum/sumsq.  One block per (b,i), thread = d.
__global__ void kedge_init(const float* __restrict__ Pn, const float* __restrict__ QR,
                           const float* __restrict__ maskf, float* etmp,
                           float* ssum, float* ssq) {
  int bi = blockIdx.x, b = bi >> 6, d = threadIdx.x;
  float Pv = Pn[(long)bi * 64 + d];
  float as = 0.f, aq = 0.f;
  long base = (long)bi * 4096;
  for (int j = 0; j < 64; ++j) {
    float v = Pv + QR[((long)b * 64 + j) * 64 + d];
    __builtin_nontemporal_store(v, &etmp[base + j * 64 + d]);   // single-use stream
    float mk = maskf[(long)bi * 64 + j];
    as += mk * v; aq += mk * v * v;
  }
  atomicAdd(&ssum[d], as);
  atomicAdd(&ssq[d], aq);
}

// masked BN finalize: e = [e +] relu((etmp - mu)/sigma);  etmp streamed non-temporally
__global__ void kmbn_fin(float* e, const float* __restrict__ etmp,
                         const float* ssum, const float* ssq, const float* cnt, int addmode) {
  long idx = (long)blockIdx.x * blockDim.x + threadIdx.x;
  if (idx >= NE) return;
  int d = (int)(idx & 63);
  float c = cnt[0];
  float mu  = ssum[d] / c;
  float var = ssq[d] / c - mu * mu;
  float x = __builtin_nontemporal_load(&etmp[idx]);
  float v = fmaxf((x - mu) * rsqrtf(var + 1e-5f), 0.f);
  e[idx] = addmode ? e[idx] + v : v;
}

// Full GatedGCN edge+node pass.  One block per (b,j), thread = d, loop over src i.
// Ce/enew are single-use streams -> non-temporal; node tensors stay cached.
__global__ void kgcn_edge(const float* __restrict__ hA, const float* __restrict__ hB,
                          const float* __restrict__ hD, const float* __restrict__ hE,
                          const float* __restrict__ Ce, const float* __restrict__ maskf,
                          float* enew, float* htmp,
                          float* hsum, float* hsq, float* esum, float* esq) {
  int bj = blockIdx.x, b = bj >> 6, j = bj & 63, d = threadIdx.x;
  float Ej = hE[(long)bj * 64 + d];
  float num = 0.f, den = 1e-6f, es = 0.f, eq = 0.f;
  for (int i = 0; i < 64; ++i) {
    long ei = (((long)b * 64 + i) * 64 + j) * 64 + d;
    float en = hD[((long)b * 64 + i) * 64 + d] + Ej + __builtin_nontemporal_load(&Ce[ei]);
    __builtin_nontemporal_store(en, &enew[ei]);
    float mk = maskf[((long)b * 64 + i) * 64 + j];
    float s = (1.f / (1.f + __expf(-en))) * mk;
    num += s * hB[((long)b * 64 + i) * 64 + d];
    den += s;
    es += mk * en; eq += mk * en * en;
  }
  float t = hA[(long)bj * 64 + d] + num / den;
  htmp[(long)bj * 64 + d] = t;
  atomicAdd(&hsum[d], t);  atomicAdd(&hsq[d], t * t);
  atomicAdd(&esum[d], es); atomicAdd(&esq[d], eq);
}

// tiny MLP readout 64 -> 32 -> 16 -> 12 (one block per graph)
__global__ void kmlp(const float* hg, const float* W1, const float* b1,
                     const float* W2, const float* b2, const float* W3, const float* b3,
                     float* out) {
  __shared__ float h0[64], h1[32], h2[16];
  int b = blockIdx.x, t = threadIdx.x;
  h0[t] = hg[b * 64 + t];
  __syncthreads();
  if (t < 32) { float s = b1[t]; for (int k = 0; k < 64; ++k) s += h0[k] * W1[k * 32 + t]; h1[t] = fmaxf(s, 0.f); }
  __syncthreads();
  if (t < 16) { float s = b2[t]; for (int k = 0; k < 32; ++k) s += h1[k] * W2[k * 16 + t]; h2[t] = fmaxf(s, 0.f); }
  __syncthreads();
  if (t < 12) { float s = b3[t]; for (int k = 0; k < 16; ++k) s += h2[k] * W3[k * 12 + t]; out[b * 12 + t] = s; }
}

// ---------------------------------------------------------------------------
static inline dim3 gs(long n, int b = 256) { return dim3((unsigned)((n + b - 1) / b)); }

// shared CrossTransformer encoder tail: out = x + LN2(relu([x, LN1(msg@Wmg)]@Wm1)@Wm2)
static void encoder_tail(hipStream_t st, const float* x, const float* msg, float* out,
                         const float* Wmg, const float* Wm1, const float* Wm2,
                         const float* ln1g, const float* ln1b,
                         const float* ln2g, const float* ln2b,
                         float* tmsg, float* msgw, float* cat, float* t128, float* m2) {
  gemm<0, 0, 0, 16>(st, msg, Wmg, nullptr, tmsg, BN, 64);
  kln<<<gs(BN), 256, 0, st>>>(tmsg, ln1g, ln1b, nullptr, msgw, BN);
  kconcat<<<gs((long)BN * 128), 256, 0, st>>>(x, msgw, cat, BN);
  gemm<0, 0, 1, 32>(st, cat, Wm1, nullptr, t128, BN, 128);          // relu, K=128
  gemm<0, 0, 0, 32>(st, t128, Wm2, nullptr, m2, BN, 64);            // K=128
  kln<<<gs(BN), 256, 0, st>>>(m2, ln2g, ln2b, x, out, BN);
}

extern "C" void kernel_launch(void* const* d_in, const int* in_sizes, int n_in,
                              void* d_out, int out_size, void* d_ws, size_t ws_size,
                              hipStream_t stream) {
  (void)in_sizes; (void)n_in; (void)out_size; (void)ws_size;
  auto f = [&](int i) { return (const float*)d_in[i]; };
  const float* feat = f(0);
  const int*   adj  = (const int*)d_in[1];
  const float *Wg1 = f(2),  *bg1 = f(3),  *Wg2 = f(4),  *bg2 = f(5);
  const float *Wpg = f(6),  *bpg = f(7),  *Wemb = f(8), *bemb = f(9);
  const float *cw  = f(10), *cb  = f(11), *Wp2 = f(12), *bp2 = f(13);
  const float *Wp3 = f(14), *bp3 = f(15), *Wp4 = f(16), *bp4 = f(17);
  const float *Wq = f(78), *Wk = f(79), *Wv = f(80), *Wmg = f(81);
  const float *Wm1 = f(82), *Wm2 = f(83);
  const float *ln1g = f(84), *ln1b = f(85), *ln2g = f(86), *ln2b = f(87);
  const float *Wr1 = f(88), *br1 = f(89), *Wr2 = f(90), *br2 = f(91), *Wr3 = f(92), *br3 = f(93);

  float* W    = (float*)d_ws;
  float* e    = W;                  // persistent edge features [B,N,N,D] (L2-resident)
  float* buf1 = W + NE;             // t1 (GTP) / Ce (layers)
  float* buf2 = W + 2 * NE;         // etmp / enew
  float* S    = W + 3 * NE;
  float* x2   = S;                 float* y1   = S + 1 * ND;  float* mmv  = S + 2 * ND;
  float* maskf= S + 3 * ND;        float* h    = S + 4 * ND;  float* hs   = S + 5 * ND;
  float* hs2g = S + 6 * ND;        float* hA   = S + 7 * ND;  float* hB   = S + 8 * ND;
  float* hD   = S + 9 * ND;        float* hE   = S + 10 * ND; float* convS= S + 11 * ND;
  float* convD= S + 12 * ND;       float* Pn   = S + 13 * ND; float* Qn   = S + 14 * ND;
  float* Rn   = S + 15 * ND;       float* QR   = S + 16 * ND; float* qq   = S + 17 * ND;
  float* msg  = S + 18 * ND;       float* tmsg = S + 19 * ND; float* msgw = S + 20 * ND;
  float* m2   = S + 21 * ND;       float* qfea = S + 22 * ND; float* edgej= S + 23 * ND;
  float* k2   = S + 24 * ND;       float* v2   = S + 25 * ND; float* htmp = S + 26 * ND;
  float* cat  = S + 27 * ND;       float* t128 = S + 29 * ND; float* KV   = S + 31 * ND;
  float* tiny = S + 32 * ND;
  float* Kb   = tiny;          float* vb   = tiny + 4096; float* Ksum = tiny + 8192;
  float* gg   = tiny + 12288;  float* hg   = tiny + 16384;
  float* econ = tiny + 20480;  float* cCv  = econ + 64;
  float* stH  = cCv + 64;      float* stHq = stH + 64;
  float* stE  = stH + 128;     float* stEq = stH + 192;
  float* cnt  = stH + 256;

  // ---------------- GTP: learned long-range edges ----------------
  gemm<0, 0, 0, 16>(stream, feat, Wg1, bg1, buf1, BN, 4096);        // t1 [4096,4096]
  kbn_nodes<<<gs((long)BB * 4096), 256, 0, stream>>>(buf1, 4096, BB * 4096);
  gemm<0, 0, 0, 16>(stream, feat, Wg2, bg2, x2, BN, 64);
  kbn_nodes<<<gs(BB * 64), 256, 0, stream>>>(x2, 64, BB * 64);
  ky1<<<gs(ND), 256, 0, stream>>>(buf1, Wpg, y1);
  gemm<0, 1, 0, 16>(stream, y1, x2, nullptr, mmv, 64, 64,           // mm[b] = y1 @ x2^T
                    BB, 64 * 64, 64 * 64, 64 * 64);
  kgtp_mask<<<BB, 64, 0, stream>>>(mmv, bpg, adj, maskf);
  kzero<<<1, 32, 0, stream>>>(cnt, 1);
  kcount<<<gs((long)BB * NN * NN), 256, 0, stream>>>(maskf, cnt, BB * NN * NN);

  // ---------------- node embedding + MERG per-node terms ----------------
  gemm<0, 0, 0, 16>(stream, feat, Wemb, bemb, h, BN, 64);
  gemm<0, 0, 0, 16>(stream, feat, Wp3, bp3, hs, BN, 64);
  kconv<<<gs(ND), 256, 0, stream>>>(feat, cw, convS, convD);
  gemm<0, 0, 0, 16>(stream, convS, Wp2, nullptr, Pn, BN, 64);
  gemm<0, 0, 0, 16>(stream, convD, Wp2, nullptr, Qn, BN, 64);
  keconst<<<1, 64, 0, stream>>>(Wp2, bp2, cb, econ);

  // ---------------- MERG global branch: g0,g1 (e == ones) ----------------
  kcopy<<<gs(ND), 256, 0, stream>>>(hs, hs2g, ND);
  const int gbase[2] = {18, 28};
  for (int li = 0; li < 2; ++li) {
    int p0 = gbase[li];
    gemm<0, 0, 0, 16>(stream, hs2g, f(p0 + 0), f(p0 + 1), hA, BN, 64);
    gemm<0, 0, 0, 16>(stream, hs2g, f(p0 + 2), f(p0 + 3), hB, BN, 64);
    kcc<<<1, 64, 0, stream>>>(f(p0 + 4), f(p0 + 5), cCv);
    gemm<0, 0, 0, 16>(stream, hs2g, f(p0 + 6), f(p0 + 7), hD, BN, 64);
    gemm<0, 0, 0, 16>(stream, hs2g, f(p0 + 8), f(p0 + 9), hE, BN, 64);
    kzero<<<1, 128, 0, stream>>>(stH, 128);
    kgcn_light<<<BN, 64, 0, stream>>>(hA, hB, hD, hE, cCv, maskf, htmp, stH, stHq);
    kbn_node_fin<<<gs(ND), 256, 0, stream>>>(hs2g, htmp, stH, stHq, 1.f / BN);
  }
  ksum_nodes<<<BB, 64, 0, stream>>>(hs2g, gg, 1.f / 64.f);          // global_g

  // ---------------- FAM (src = broadcast global) ----------------
  gemm<0, 0, 0, 16>(stream, hs, Wq, nullptr, qq, BN, 64);
  gemm<0, 0, 2, 16>(stream, gg, Wk, nullptr, Kb, 64, 64);           // elu+1
  gemm<0, 0, 0, 16>(stream, gg, Wv, nullptr, vb, 64, 64);
  kfam_msg<<<gs(BN), 256, 0, stream>>>(qq, Kb, vb, msg);
  encoder_tail(stream, hs, msg, qfea, Wmg, Wm1, Wm2, ln1g, ln1b, ln2g, ln2b,
               tmsg, msgw, cat, t128, m2);

  // ---------------- ARM (pairwise encoder, collapsed to per-node) ----------------
  gemm<0, 0, 0, 16>(stream, qfea, Wq, nullptr, qq, BN, 64);
  gemm<0, 0, 2, 16>(stream, qfea, Wk, nullptr, k2, BN, 64);         // elu+1
  gemm<0, 0, 0, 16>(stream, qfea, Wv, nullptr, v2, BN, 64);
  gemm<1, 0, 0, 16>(stream, k2, v2, nullptr, KV, 64, 64,            // KV = K^T V / 64
                    BB, 64 * 64, 64 * 64, 64 * 64, 1.f / 64.f);
  ksum_nodes<<<BB, 64, 0, stream>>>(k2, Ksum, 64.f);                // Ksum over 4096 rows
  karm_msg<<<BN, 64, 0, stream>>>(qq, Ksum, KV, msg);
  encoder_tail(stream, qfea, msg, edgej, Wmg, Wm1, Wm2, ln1g, ln1b, ln2g, ln2b,
               tmsg, msgw, cat, t128, m2);
  gemm<0, 0, 0, 16>(stream, edgej, Wp4, bp4, Rn, BN, 64);           // lr_e_global row (j)

  // ---------------- initial edge features + masked BN ----------------
  kaddvec3<<<gs(ND), 256, 0, stream>>>(Qn, Rn, econ, QR);
  kzero<<<1, 128, 0, stream>>>(stE, 128);
  kedge_init<<<BN, 64, 0, stream>>>(Pn, QR, maskf, buf2, stE, stEq);
  kmbn_fin<<<gs(NE), 256, 0, stream>>>(e, buf2, stE, stEq, cnt, 0);

  // ---------------- main GatedGCN layers m0..m3 ----------------
  const int mbase[4] = {38, 48, 58, 68};
  for (int li = 0; li < 4; ++li) {
    int p0 = mbase[li];
    gemm<0, 0, 0, 16>(stream, h, f(p0 + 0), f(p0 + 1), hA, BN, 64);
    gemm<0, 0, 0, 16>(stream, h, f(p0 + 2), f(p0 + 3), hB, BN, 64);
    gemm<0, 0, 0, 16>(stream, h, f(p0 + 6), f(p0 + 7), hD, BN, 64);
    gemm<0, 0, 0, 16>(stream, h, f(p0 + 8), f(p0 + 9), hE, BN, 64);
    gemm<0, 0, 0, 16, /*NT store*/1>(stream, e, f(p0 + 4), f(p0 + 5), buf1,
                                     NE / 64, 64);                  // Ce: M = 262144
    kzero<<<1, 256, 0, stream>>>(stH, 256);
    kgcn_edge<<<BN, 64, 0, stream>>>(hA, hB, hD, hE, buf1, maskf, buf2, htmp,
                                     stH, stHq, stE, stEq);
    kbn_node_fin<<<gs(ND), 256, 0, stream>>>(h, htmp, stH, stHq, 1.f / BN);
    kmbn_fin<<<gs(NE), 256, 0, stream>>>(e, buf2, stE, stEq, cnt, 1);
  }

  // ---------------- readout ----------------
  ksum_nodes<<<BB, 64, 0, stream>>>(h, hg, 1.f / 64.f);
  kmlp<<<BB, 64, 0, stream>>>(hg, Wr1, br1, Wr2, br2, Wr3, br3, (float*)d_out);
}